// GraphSAGE_35253091565752
// MI455X (gfx1250) — compile-verified
//
#include <hip/hip_runtime.h>
#include <hip/hip_bf16.h>

typedef __attribute__((ext_vector_type(2))) float v2f;
typedef __attribute__((ext_vector_type(8))) float v8f;

#define D_FEAT 128

// ---------------------------------------------------------------------------
// Zero a float region (grid-stride).
// ---------------------------------------------------------------------------
__global__ void zero_kernel(float* __restrict__ p, size_t n) {
    size_t i = (size_t)blockIdx.x * blockDim.x + threadIdx.x;
    size_t stride = (size_t)gridDim.x * blockDim.x;
    for (; i < n; i += stride) p[i] = 0.0f;
}

// ---------------------------------------------------------------------------
// In-degree: deg[dst] += 1 per edge.
// ---------------------------------------------------------------------------
__global__ void deg_kernel(const int* __restrict__ dst, float* __restrict__ deg,
                           int nEdges) {
    int e = blockIdx.x * blockDim.x + threadIdx.x;
    if (e < nEdges) atomicAdd(&deg[dst[e]], 1.0f);
}

// ---------------------------------------------------------------------------
// Neighbor scatter-add: agg[dst, :] += h[src, :].
// One wave32 per edge; each lane handles 4 consecutive floats (float4 load,
// 4x global_atomic_add_f32 into the L2-resident agg buffer).
// ---------------------------------------------------------------------------
__global__ __launch_bounds__(256)
void scatter_kernel(const int* __restrict__ src, const int* __restrict__ dst,
                    const float* __restrict__ h, float* __restrict__ agg,
                    int nEdges) {
    int edge = blockIdx.x * 8 + (threadIdx.x >> 5);
    if (edge >= nEdges) return;
    int lane = threadIdx.x & 31;
    int s = src[edge];
    int d = dst[edge];
    const float4 v =
        *reinterpret_cast<const float4*>(h + (size_t)s * D_FEAT + lane * 4);
    float* p = agg + (size_t)d * D_FEAT + lane * 4;
    atomicAdd(p + 0, v.x);
    atomicAdd(p + 1, v.y);
    atomicAdd(p + 2, v.z);
    atomicAdd(p + 3, v.w);
}

// ---------------------------------------------------------------------------
// Fused normalize + GEMM (+bias, optional ReLU) via fp32 WMMA.
//   out[m, n] = act( sum_k ((agg[m,k] + hin[m,k]) / (deg[m]+1)) * W[k,n] + b[n] )
// Block: 16 rows x 128 cols. 8 waves, each wave owns one 16x16 N-tile.
// A-tile (16x128 normalized fp32) staged in LDS; 32x v_wmma_f32_16x16x4_f32.
// ---------------------------------------------------------------------------
template <bool RELU>
__global__ __launch_bounds__(256)
void gemm_norm_kernel(const float* __restrict__ agg,
                      const float* __restrict__ hin,
                      const float* __restrict__ deg,
                      const float* __restrict__ W,
                      const float* __restrict__ bias,
                      float* __restrict__ out, int nNodes) {
    __shared__ float As[16 * D_FEAT];  // 8 KB

    const int mBase = blockIdx.x * 16;
    const int tid = threadIdx.x;

    // Cooperative load + fused normalization: 2048 elements / 256 threads.
    for (int i = tid; i < 16 * D_FEAT; i += 256) {
        int r = i >> 7;          // row in tile
        int cc = i & (D_FEAT - 1);
        int node = mBase + r;
        if (node >= nNodes) node = nNodes - 1;  // tail clamp (unused: 100000%16==0)
        size_t idx = (size_t)node * D_FEAT + cc;
        float rden = 1.0f / (deg[node] + 1.0f);
        As[i] = (agg[idx] + hin[idx]) * rden;
    }
    __syncthreads();

    const int wave = tid >> 5;            // 0..7 -> N-tile
    const int lane = tid & 31;
    const int colBase = wave * 16;
    const int col = colBase + (lane & 15);
    const int khalf = (lane >> 4) * 2;    // lanes 0-15: K{0,1}; lanes 16-31: K{2,3}
    const int arow = lane & 15;

    v8f c = {};
#pragma unroll
    for (int k0 = 0; k0 < D_FEAT; k0 += 4) {
        v2f a, b;
        // A 16x4 fragment (ISA layout: VGPR0 = K even, VGPR1 = K odd)
        a.x = As[arow * D_FEAT + k0 + khalf];
        a.y = As[arow * D_FEAT + k0 + khalf + 1];
        // B 4x16 fragment from W[k, n] (row-major [K x N]); coalesced per half-wave
        b.x = W[(size_t)(k0 + khalf) * D_FEAT + col];
        b.y = W[(size_t)(k0 + khalf + 1) * D_FEAT + col];
        // (neg_a, A, neg_b, B, c_mod, C, reuse_a, reuse_b)
        c = __builtin_amdgcn_wmma_f32_16x16x4_f32(false, a, false, b, (short)0,
                                                  c, false, false);
    }

    const float bv = bias[col];
    const int rowTop = mBase + (lane >> 4) * 8;  // D layout: VGPR j -> M=j / M=j+8
#pragma unroll
    for (int j = 0; j < 8; ++j) {
        int row = rowTop + j;
        if (row < nNodes) {
            float v = c[j] + bv;
            if (RELU) v = fmaxf(v, 0.0f);
            out[(size_t)row * D_FEAT + col] = v;
        }
    }
}

// ---------------------------------------------------------------------------
// Inputs (setup_inputs order): x[N*128] f32, edge_index[2*E] i32 (JAX x64 off),
// W1[128*128], b1[128], W2[128*128], b2[128]. Output: [N*128] f32.
// ---------------------------------------------------------------------------
extern "C" void kernel_launch(void* const* d_in, const int* in_sizes, int n_in,
                              void* d_out, int out_size, void* d_ws,
                              size_t ws_size, hipStream_t stream) {
    const float* x = (const float*)d_in[0];
    const int* eidx = (const int*)d_in[1];
    const float* W1 = (const float*)d_in[2];
    const float* b1 = (const float*)d_in[3];
    const float* W2 = (const float*)d_in[4];
    const float* b2 = (const float*)d_in[5];
    float* out = (float*)d_out;

    const int nNodes = in_sizes[0] / D_FEAT;
    const int nEdges = in_sizes[1] / 2;
    const int* src = eidx;            // edge_index[0]
    const int* dst = eidx + nEdges;   // edge_index[1]

    // Workspace layout: deg | agg | h1   (deg and agg contiguous -> one zero pass)
    float* deg = (float*)d_ws;
    float* agg = deg + nNodes;                       // nNodes*128 floats
    float* h1 = agg + (size_t)nNodes * D_FEAT;       // nNodes*128 floats

    const size_t aggN = (size_t)nNodes * D_FEAT;
    const int zeroBlocks = 4096;

    // Layer 1 ---------------------------------------------------------------
    zero_kernel<<<zeroBlocks, 256, 0, stream>>>(deg, (size_t)nNodes + aggN);
    deg_kernel<<<(nEdges + 255) / 256, 256, 0, stream>>>(dst, deg, nEdges);
    scatter_kernel<<<(nEdges + 7) / 8, 256, 0, stream>>>(src, dst, x, agg, nEdges);
    gemm_norm_kernel<true><<<(nNodes + 15) / 16, 256, 0, stream>>>(
        agg, x, deg, W1, b1, h1, nNodes);

    // Layer 2 ---------------------------------------------------------------
    zero_kernel<<<zeroBlocks, 256, 0, stream>>>(agg, aggN);
    scatter_kernel<<<(nEdges + 7) / 8, 256, 0, stream>>>(src, dst, h1, agg, nEdges);
    gemm_norm_kernel<false><<<(nNodes + 15) / 16, 256, 0, stream>>>(
        agg, h1, deg, W2, b2, out, nNodes);
}